// NGCF_670014898468
// MI455X (gfx1250) — compile-verified
//
#include <hip/hip_runtime.h>
#include <hip/hip_bf16.h>
#include <math.h>

// ---------------------------------------------------------------------------
// NGCF on MI455X (gfx1250, wave32).
//  - f32 WMMA (V_WMMA_F32_16X16X4_F32) for the [N,64]x[64,64] GEMMs
//  - wave-per-edge float2 scatter with global f32 atomics (agg fits in 192MB L2)
//  - wave-per-node leaky-relu + L2 norm (wave32 shfl reduction)
//  - wave-per-interaction final dot products over the 4 embedding tables
// ---------------------------------------------------------------------------

typedef __attribute__((ext_vector_type(2))) float v2f;
typedef __attribute__((ext_vector_type(8))) float v8f;

#define D 64
#define SLOPE 0.2f

// ---------------------------------------------------------------- degree ----
__global__ void zero_deg_kernel(float* __restrict__ deg, int n) {
    int i = blockIdx.x * blockDim.x + threadIdx.x;
    if (i < n) deg[i] = 0.0f;
}

__global__ void deg_atomic_kernel(const int* __restrict__ users,
                                  const int* __restrict__ pos,
                                  float* __restrict__ deg, int b) {
    int i = blockIdx.x * blockDim.x + threadIdx.x;
    if (i < b) {
        // col = concat(pos, users): each interaction bumps both endpoints
        atomicAdd(&deg[pos[i]], 1.0f);
        atomicAdd(&deg[users[i]], 1.0f);
    }
}

__global__ void dis_kernel(const float* __restrict__ deg,
                           float* __restrict__ dis,
                           float* __restrict__ dis_sl, int n) {
    int i = blockIdx.x * blockDim.x + threadIdx.x;
    if (i < n) {
        float dv = deg[i];
        dis[i]    = dv > 0.0f ? rsqrtf(dv) : 0.0f;
        dis_sl[i] = rsqrtf(dv + 1.0f);
    }
}

// ------------------------------------------------------------------ GEMM ----
// x1 = emb @ W1^T, x2 = emb @ W2^T.  One wave per 16-row tile; W1/W2 in LDS.
// A frag (16x4 f32): lane l: M=l&15, VGPR0=K(2*hi), VGPR1=K(2*hi+1), hi=l>>4.
// B frag (4x16 = W^T tile): symmetric layout -> reads W row-major pairs.
// C (16x16 f32): VGPR r, lanes<16 -> (M=r,N=lane); lanes>=16 -> (M=r+8,N=l-16).
__global__ void __launch_bounds__(256)
gemm_x1x2_kernel(const float* __restrict__ emb,
                 const float* __restrict__ W1,
                 const float* __restrict__ W2,
                 float* __restrict__ x1,
                 float* __restrict__ x2, int n) {
    __shared__ float sW[2 * D * D];   // 32 KB
    for (int i = threadIdx.x; i < D * D; i += 256) {
        sW[i]         = W1[i];
        sW[D * D + i] = W2[i];
    }
    __syncthreads();

    const int wave = threadIdx.x >> 5;
    const int lane = threadIdx.x & 31;
    const int rt   = blockIdx.x * 8 + wave;       // wave-uniform
    const int nrt  = n >> 4;
    if (rt < nrt) {
        const int r0 = rt << 4;
        const int m  = lane & 15;
        const int hi = lane >> 4;

        // A fragments for the whole K=64 (16 steps of K=4), reused 8x.
        v2f a[16];
        const float* arow = emb + (size_t)(r0 + m) * D + 2 * hi;
#pragma unroll
        for (int ks = 0; ks < 16; ++ks)
            a[ks] = *(const v2f*)(arow + ks * 4);

#pragma unroll
        for (int w = 0; w < 2; ++w) {
            const float* Ws   = sW + w * D * D;
            float*       xout = w ? x2 : x1;
#pragma unroll
            for (int ct = 0; ct < 4; ++ct) {
                v8f c = {};
                const float* brow = Ws + (size_t)(ct * 16 + m) * D + 2 * hi;
#pragma unroll
                for (int ks = 0; ks < 16; ++ks) {
                    v2f b = *(const v2f*)(brow + ks * 4);   // ds_load_b64
                    c = __builtin_amdgcn_wmma_f32_16x16x4_f32(
                        false, a[ks], false, b, (short)0, c, false, false);
                }
                float* orow = xout + (size_t)(r0 + 8 * hi) * D + ct * 16 + m;
#pragma unroll
                for (int r = 0; r < 8; ++r)
                    orow[(size_t)r * D] = c[r];
            }
        }
    }
}

// ---------------------------------------------------- self-loop agg init ----
// agg[i] = dis_sl[i]^2 * x1[i]   (norm1e == 0 on self loops)
__global__ void init_agg_kernel(const float* __restrict__ x1,
                                const float* __restrict__ dis_sl,
                                float* __restrict__ agg, int n) {
    int idx = blockIdx.x * blockDim.x + threadIdx.x;   // n*32 threads, 2 f32 each
    if (idx < n * 32) {
        int node = idx >> 5;
        int d    = (idx & 31) * 2;
        float s  = dis_sl[node];
        s        = s * s;
        v2f xv   = *(const v2f*)(x1 + (size_t)node * D + d);
        v2f o; o.x = s * xv.x; o.y = s * xv.y;
        *(v2f*)(agg + (size_t)node * D + d) = o;
    }
}

// -------------------------------------------------------- edge scatter ------
// m = norm2 * x1[row] + norm1 * (emb[row] * x2[col]); agg[col] += m
__global__ void scatter_kernel(const int* __restrict__ users,
                               const int* __restrict__ pos,
                               const float* __restrict__ emb,
                               const float* __restrict__ x1,
                               const float* __restrict__ x2,
                               const float* __restrict__ dis,
                               const float* __restrict__ dis_sl,
                               float* __restrict__ agg, int b) {
    int e    = blockIdx.x * 8 + (threadIdx.x >> 5);   // wave per edge
    int lane = threadIdx.x & 31;
    int e2   = 2 * b;
    if (e < e2) {
        int row, col;
        if (e < b) { row = users[e];     col = pos[e]; }
        else       { row = pos[e - b];   col = users[e - b]; }
        float norm1 = dis[row]    * dis[col];
        float norm2 = dis_sl[row] * dis_sl[col];
        int d = lane * 2;
        v2f x1r = *(const v2f*)(x1  + (size_t)row * D + d);
        v2f er  = *(const v2f*)(emb + (size_t)row * D + d);
        v2f x2c = *(const v2f*)(x2  + (size_t)col * D + d);
        float mx = norm2 * x1r.x + norm1 * (er.x * x2c.x);
        float my = norm2 * x1r.y + norm1 * (er.y * x2c.y);
        atomicAdd(&agg[(size_t)col * D + d],     mx);
        atomicAdd(&agg[(size_t)col * D + d + 1], my);
    }
}

// ------------------------------------------- leaky relu + L2 normalize ------
__global__ void finalize_kernel(float* __restrict__ agg, int n) {
    int node = blockIdx.x * 8 + (threadIdx.x >> 5);   // wave per node
    int lane = threadIdx.x & 31;
    if (node < n) {
        int d  = lane * 2;
        v2f v  = *(const v2f*)(agg + (size_t)node * D + d);
        v.x = v.x > 0.0f ? v.x : SLOPE * v.x;
        v.y = v.y > 0.0f ? v.y : SLOPE * v.y;
        float ss = v.x * v.x + v.y * v.y;
#pragma unroll
        for (int off = 16; off > 0; off >>= 1)
            ss += __shfl_xor(ss, off, 32);
        float inv = 1.0f / fmaxf(sqrtf(ss), 1e-12f);
        v.x *= inv; v.y *= inv;
        *(v2f*)(agg + (size_t)node * D + d) = v;
    }
}

// --------------------------------------------------------- similarities -----
__global__ void sim_kernel(const int* __restrict__ users,
                           const int* __restrict__ pos,
                           const int* __restrict__ neg,
                           const float* __restrict__ E,
                           const float* __restrict__ e1,
                           const float* __restrict__ e2,
                           const float* __restrict__ e3,
                           float* __restrict__ out, int b) {
    int i    = blockIdx.x * 8 + (threadIdx.x >> 5);   // wave per interaction
    int lane = threadIdx.x & 31;
    if (i < b) {
        int u = users[i], p = pos[i], g = neg[i];
        int d = lane * 2;
        const float* bufs[4] = {E, e1, e2, e3};
        float ps = 0.0f, ns = 0.0f;
#pragma unroll
        for (int t = 0; t < 4; ++t) {
            const float* bf = bufs[t];
            v2f uv = *(const v2f*)(bf + (size_t)u * D + d);
            v2f pv = *(const v2f*)(bf + (size_t)p * D + d);
            v2f gv = *(const v2f*)(bf + (size_t)g * D + d);
            ps += uv.x * pv.x + uv.y * pv.y;
            ns += uv.x * gv.x + uv.y * gv.y;
        }
#pragma unroll
        for (int off = 16; off > 0; off >>= 1) {
            ps += __shfl_xor(ps, off, 32);
            ns += __shfl_xor(ns, off, 32);
        }
        if (lane == 0) {
            out[i]     = ps;
            out[b + i] = ns;
        }
    }
}

// ------------------------------------------------------------------ host ----
extern "C" void kernel_launch(void* const* d_in, const int* in_sizes, int n_in,
                              void* d_out, int out_size, void* d_ws, size_t ws_size,
                              hipStream_t stream) {
    const float* E   = (const float*)d_in[0];
    const float* W1a = (const float*)d_in[1];
    const float* W2a = (const float*)d_in[2];
    const float* W1b = (const float*)d_in[3];
    const float* W2b = (const float*)d_in[4];
    const float* W1c = (const float*)d_in[5];
    const float* W2c = (const float*)d_in[6];
    const int* users = (const int*)d_in[7];
    const int* pos   = (const int*)d_in[8];
    const int* neg   = (const int*)d_in[9];
    float* out = (float*)d_out;

    const int n = in_sizes[0] / D;     // 500,000 nodes
    const int b = in_sizes[7];         // 2,000,000 interactions

    // workspace layout (floats): deg | dis | dis_sl | x1 | x2 | e1 | e2 | e3
    float* ws     = (float*)d_ws;
    float* deg    = ws;
    float* dis    = deg    + n;
    float* dis_sl = dis    + n;
    float* x1     = dis_sl + n;
    float* x2     = x1 + (size_t)n * D;
    float* e1     = x2 + (size_t)n * D;
    float* e2     = e1 + (size_t)n * D;
    float* e3     = e2 + (size_t)n * D;

    const int T = 256;
    // --- degrees and symmetric norms ---
    zero_deg_kernel<<<(n + T - 1) / T, T, 0, stream>>>(deg, n);
    deg_atomic_kernel<<<(b + T - 1) / T, T, 0, stream>>>(users, pos, deg, b);
    dis_kernel<<<(n + T - 1) / T, T, 0, stream>>>(deg, dis, dis_sl, n);

    const int nrt        = n >> 4;                 // 16-row tiles
    const int gemm_grid  = (nrt + 7) / 8;          // 8 waves/block
    const int nodeElems  = n * 32;                 // threads for float2 layout
    const int edge_grid  = (2 * b + 7) / 8;        // wave per edge
    const int node_wgrid = (n + 7) / 8;            // wave per node

    const float* W1s[3] = {W1a, W1b, W1c};
    const float* W2s[3] = {W2a, W2b, W2c};
    float*       eOut[3] = {e1, e2, e3};

    const float* embIn = E;
    for (int l = 0; l < 3; ++l) {
        float* agg = eOut[l];
        gemm_x1x2_kernel<<<gemm_grid, T, 0, stream>>>(embIn, W1s[l], W2s[l], x1, x2, n);
        init_agg_kernel<<<(nodeElems + T - 1) / T, T, 0, stream>>>(x1, dis_sl, agg, n);
        scatter_kernel<<<edge_grid, T, 0, stream>>>(users, pos, embIn, x1, x2,
                                                    dis, dis_sl, agg, b);
        finalize_kernel<<<node_wgrid, T, 0, stream>>>(agg, n);
        embIn = agg;
    }

    sim_kernel<<<(b + 7) / 8, T, 0, stream>>>(users, pos, neg, E, e1, e2, e3, out, b);
}